// LinearGaussianQ_1202590843622
// MI455X (gfx1250) — compile-verified
//
#include <hip/hip_runtime.h>
#include <math.h>

#define DZV 16
#define DXV 16
#define TT 256
#define LOG2PI 1.8378770664093453f

typedef __attribute__((ext_vector_type(2))) float v2f;
typedef __attribute__((ext_vector_type(8))) float v8f;

// ---------- 256-thread cooperative 16x16 helpers (LDS tiles) ----------
__device__ __forceinline__ void mm_nn(float* C, const float* A, const float* B, int tid){
  int i = tid >> 4, j = tid & 15;
  float a = 0.f;
  #pragma unroll
  for (int k = 0; k < 16; ++k) a += A[i*16+k] * B[k*16+j];
  C[i*16+j] = a;
  __syncthreads();
}
__device__ __forceinline__ void mm_tn(float* C, const float* A, const float* B, int tid){
  int i = tid >> 4, j = tid & 15;
  float a = 0.f;
  #pragma unroll
  for (int k = 0; k < 16; ++k) a += A[k*16+i] * B[k*16+j];
  C[i*16+j] = a;
  __syncthreads();
}
__device__ __forceinline__ void mm_nt(float* C, const float* A, const float* B, int tid){
  int i = tid >> 4, j = tid & 15;
  float a = 0.f;
  #pragma unroll
  for (int k = 0; k < 16; ++k) a += A[i*16+k] * B[j*16+k];
  C[i*16+j] = a;
  __syncthreads();
}

// SPD inverse via Gauss-Jordan (no pivoting; SPD pivots stay > 0).
// 256 threads: row i = tid>>4, column pair (2cp,2cp+1) of the 32-wide [A|I].
// Optionally accumulates log det(A) into *logdet (caller zeroes it).
__device__ __forceinline__ void spd_invert(float* Ainv, const float* A, float* logdet,
                                           int tid, float* aug){
  const int i = tid >> 4, cp = tid & 15;
  const int c0 = 2*cp, c1 = 2*cp + 1;
  aug[i*33+c0] = (c0 < 16) ? A[i*16+c0] : ((c0-16)==i ? 1.f : 0.f);
  aug[i*33+c1] = (c1 < 16) ? A[i*16+c1] : ((c1-16)==i ? 1.f : 0.f);
  __syncthreads();
  for (int k = 0; k < 16; ++k){
    float piv = aug[k*33+k];
    float fac = aug[i*33+k];
    float rk0 = aug[k*33+c0];
    float rk1 = aug[k*33+c1];
    __syncthreads();
    float invp = 1.f / piv;
    if (i == k){
      aug[i*33+c0] = rk0 * invp;
      aug[i*33+c1] = rk1 * invp;
      if (cp == 0 && logdet) *logdet += logf(piv);
    } else {
      float f = fac * invp;
      aug[i*33+c0] -= f * rk0;
      aug[i*33+c1] -= f * rk1;
    }
    __syncthreads();
  }
  if (c0 >= 16) Ainv[i*16 + (c0-16)] = aug[i*33+c0];
  if (c1 >= 16) Ainv[i*16 + (c1-16)] = aug[i*33+c1];
  __syncthreads();
}

// ---------- WMMA f32 16x16x16 via 4 chained V_WMMA_F32_16X16X4_F32 ----------
// A frag (16x4 f32): lane m=lane&15; VGPR j holds K = 4*slice + 2*(lane>>4) + j
__device__ __forceinline__ void load_afrag_g(v2f a[4], const float* M, int lane){
  int m = lane & 15, kh = (lane >> 4) << 1;
  #pragma unroll
  for (int s = 0; s < 4; ++s)
    a[s] = *(const v2f*)(M + m*16 + 4*s + kh);
}
__device__ __forceinline__ void load_afrag_s(v2f a[4], const float* M, int lane, int ld){
  int m = lane & 15, kh = (lane >> 4) << 1;
  #pragma unroll
  for (int s = 0; s < 4; ++s)
    a[s] = (v2f){ M[m*ld + 4*s + kh], M[m*ld + 4*s + kh + 1] };
}
// B frag (4x16 f32): lane n=lane&15; VGPR j holds K = 4*slice + 2*(lane>>4) + j
__device__ __forceinline__ void load_bfrag_s(v2f b[4], const float* M, int lane, int ld){
  int n = lane & 15, kh = (lane >> 4) << 1;
  #pragma unroll
  for (int s = 0; s < 4; ++s)
    b[s] = (v2f){ M[(4*s+kh)*ld + n], M[(4*s+kh+1)*ld + n] };
}
__device__ __forceinline__ v8f mma16(const v2f a[4], const v2f b[4]){
  v8f c = {0.f,0.f,0.f,0.f,0.f,0.f,0.f,0.f};
  #pragma unroll
  for (int s = 0; s < 4; ++s)
    c = __builtin_amdgcn_wmma_f32_16x16x4_f32(false, a[s], false, b[s],
                                              (short)0, c, false, false);
  return c;
}

// One 16x16 history matrix M (global, row-major):
//   Y = M @ Bm          (WMMA)
//   Z = Om @ Y          (WMMA, via LDS re-fragmentation of Y)
//   ret = sum(Z .* M) (transDot=false)  or  sum(Z .* M^T) (transDot=true)
//   if doUpdate: M <- Y @ ct  (== M @ bA since bA = bcov @ ct)
__device__ __forceinline__ float wave_process(float* Mg, const float* Om, const float* Bm,
                                              const v2f ctB[4], float* stg, int lane,
                                              bool transDot, bool doUpdate){
  v2f a[4];  load_afrag_g(a, Mg, lane);
  v2f bb[4]; load_bfrag_s(bb, Bm, lane, 16);
  v8f Y = mma16(a, bb);
  const int n = lane & 15, hi = lane >> 4;
  #pragma unroll
  for (int r = 0; r < 8; ++r) stg[(r + 8*hi)*18 + n] = Y[r];   // C layout -> row-major tile
  asm volatile("" ::: "memory");   // same-wave DS ops are in order; block compiler reordering
  v2f yb[4]; load_bfrag_s(yb, stg, lane, 18);
  v2f om[4]; load_afrag_s(om, Om, lane, 16);
  v8f Z = mma16(om, yb);
  float acc = 0.f;
  #pragma unroll
  for (int r = 0; r < 8; ++r){
    int rr = r + 8*hi;
    float mval = transDot ? Mg[n*16 + rr] : Mg[rr*16 + n];
    acc += Z[r] * mval;
  }
  if (doUpdate){
    v2f ya[4]; load_afrag_s(ya, stg, lane, 18);
    v8f Nw = mma16(ya, ctB);
    #pragma unroll
    for (int r = 0; r < 8; ++r) Mg[(r + 8*hi)*16 + n] = Nw[r];
  }
  return acc;
}

__global__ __launch_bounds__(256) void lgq_kernel(
    const float* __restrict__ obs,
    const float* __restrict__ g_p_prior_mean, const float* __restrict__ g_p_prior_cov,
    const float* __restrict__ g_p_trans_w,  const float* __restrict__ g_p_trans_b,
    const float* __restrict__ g_p_trans_cov,
    const float* __restrict__ g_p_emis_w,   const float* __restrict__ g_p_emis_b,
    const float* __restrict__ g_p_emis_cov,
    const float* __restrict__ g_q_prior_mean, const float* __restrict__ g_q_prior_cov,
    const float* __restrict__ g_q_trans_w,  const float* __restrict__ g_q_trans_b,
    const float* __restrict__ g_q_trans_cov,
    const float* __restrict__ g_q_emis_w,   const float* __restrict__ g_q_emis_b,
    const float* __restrict__ g_q_emis_cov,
    float* __restrict__ out, float* __restrict__ ws)
{
  __shared__ float qW[256], qcov[256], Hq[256], Rq[256], pW[256], pH[256];
  __shared__ float qb[16], hb[16], pb[16], peb[16];
  __shared__ float qprec[256], ct_m[256], CtW[256], Mw[256];
  __shared__ float OmT[256], Om0[256], OmO[256];
  __shared__ float fc[256], fprec[256], bcov[256], bA_m[256], pc_m[256];
  __shared__ float tmpA[256], tmpB[256];
  __shared__ float fm[16], pm_v[16], v_v[16], ba_v[16];
  __shared__ float aug[16*33];
  __shared__ float stage[8][16*18];
  __shared__ float red[256];
  __shared__ float scal[8];   // 0: GJ logdet scratch, 1: constants, 2: ld_pt, 3: ld_pe, 4: ld_pp

  const int tid  = threadIdx.x;
  const int lane = tid & 31, wave = tid >> 5;
  float* stg = stage[wave];
  float  acc = 0.f;

  float* transA = ws;               // 256 * 256
  float* obsA   = ws + 65536;       // 256 * 256
  float* transB = ws + 131072;      // 256 * 16
  float* obsB   = ws + 135168;      // 256 * 16

  // -------- load constants to LDS --------
  qW[tid] = g_q_trans_w[tid];  qcov[tid] = g_q_trans_cov[tid];
  Hq[tid] = g_q_emis_w[tid];   Rq[tid]   = g_q_emis_cov[tid];
  pW[tid] = g_p_trans_w[tid];  pH[tid]   = g_p_emis_w[tid];
  if (tid < 16){ qb[tid]=g_q_trans_b[tid]; hb[tid]=g_q_emis_b[tid];
                 pb[tid]=g_p_trans_b[tid]; peb[tid]=g_p_emis_b[tid]; }
  if (tid < 8) scal[tid] = 0.f;
  __syncthreads();

  // -------- precompute precisions / log dets / derived constants --------
  spd_invert(qprec, g_q_trans_cov, nullptr,  tid, aug);
  spd_invert(tmpA,  g_p_trans_cov, &scal[2], tid, aug);   // p_trans_prec, ld_pt
  OmT[tid] = -0.5f * tmpA[tid];
  __syncthreads();
  mm_tn(tmpB, pW, tmpA, tid);                             // pW^T * Pp
  mm_nn(Mw,   tmpB, pW, tid);                             // Mw = pW^T Pp pW
  spd_invert(tmpA, g_p_emis_cov,  &scal[3], tid, aug);    // ld_pe
  OmO[tid] = -0.5f * tmpA[tid];
  __syncthreads();
  spd_invert(tmpA, g_p_prior_cov, &scal[4], tid, aug);    // ld_pp
  Om0[tid] = -0.5f * tmpA[tid];
  __syncthreads();
  mm_tn(ct_m, qW, qprec, tid);                            // ct = qW^T qprec
  mm_nn(CtW,  ct_m, qW, tid);                             // ct qW

  // Kalman measurement update on (pm_v, pc_m) with observation y -> (fm, fc)
  auto kalman = [&](const float* y){
    mm_nn(tmpA, Hq, pc_m, tid);
    mm_nt(tmpB, tmpA, Hq, tid);
    tmpB[tid] += Rq[tid];
    __syncthreads();
    spd_invert(fprec, tmpB, nullptr, tid, aug);           // fprec <- S^-1 (scratch)
    mm_nt(tmpA, pc_m, Hq, tid);                           // U = pc Hq^T
    mm_nn(tmpB, tmpA, fprec, tid);                        // K = U S^-1
    if (tid < 16){
      float a = 0.f;
      for (int k = 0; k < 16; ++k) a += Hq[tid*16+k]*pm_v[k];
      v_v[tid] = y[tid] - a - hb[tid];
    }
    __syncthreads();
    if (tid < 16){
      float a = 0.f;
      for (int k = 0; k < 16; ++k) a += tmpB[tid*16+k]*v_v[k];
      fm[tid] = pm_v[tid] + a;
    }
    __syncthreads();
    mm_nn(tmpA, tmpB, Hq, tid);                           // KH
    { int i = tid>>4, j = tid&15;
      float a = pc_m[i*16+j];
      for (int k = 0; k < 16; ++k) a -= tmpA[i*16+k]*pc_m[k*16+j];
      fc[i*16+j] = a; }                                   // (I-KH) pc
    __syncthreads();
  };

  // -------- init: Kalman with q prior and y0; constants; history slot 0 --------
  pc_m[tid] = g_q_prior_cov[tid];
  if (tid < 16) pm_v[tid] = g_q_prior_mean[tid];
  __syncthreads();
  kalman(obs);
  if (tid == 0)
    scal[1] = -0.5f*(DZV*LOG2PI + scal[4]) - 0.5f*(DXV*LOG2PI + scal[3]);
  { int i = tid>>4, j = tid&15;
    transA[tid] = (i==j) ? 1.f : 0.f;
    obsA[tid]   = pH[tid]; }
  if (tid < 16){ transB[tid] = -g_p_prior_mean[tid];
                 obsB[tid]   = peb[tid] - obs[tid]; }
  __syncthreads();

  // -------- sequential scan --------
  for (int t = 1; t < TT; ++t){
    const float* y = obs + t*16;
    spd_invert(fprec, fc, nullptr, tid, aug);             // fprec = inv(fc_prev)
    tmpA[tid] = CtW[tid] + fprec[tid];
    if (tid == 0) scal[0] = 0.f;
    __syncthreads();
    spd_invert(bcov, tmpA, &scal[0], tid, aug);           // bcov, scal[0]=logdet(ct qW + fprec)
    mm_nn(bA_m, bcov, ct_m, tid);                         // bA
    if (tid < 16){
      float a = 0.f, b = 0.f;
      for (int k = 0; k < 16; ++k){ a += fprec[tid*16+k]*fm[k]; b += ct_m[tid*16+k]*qb[k]; }
      v_v[tid] = a - b;                                   // v = fprec fm - ct qb
    }
    __syncthreads();
    if (tid < 16){
      float a = 0.f;
      for (int k = 0; k < 16; ++k) a += bcov[tid*16+k]*v_v[k];
      ba_v[tid] = a;                                      // ba = bcov v
    }
    __syncthreads();
    if (tid < 16){
      float a = 0.f;
      for (int k = 0; k < 16; ++k) a += qW[tid*16+k]*fm[k];
      pm_v[tid] = a + qb[tid];                            // pm = qW fm + qb
    }
    __syncthreads();
    mm_nn(tmpA, qW, fc, tid);
    mm_nt(pc_m, tmpA, qW, tid);
    pc_m[tid] += qcov[tid];                               // pc = qW fc qW^T + qcov
    __syncthreads();
    if (tid == 0){                                        // per-step scalar constants
      float trm = 0.f;
      for (int q2 = 0; q2 < 256; ++q2) trm += Mw[q2]*bcov[q2];
      scal[1] += -0.5f*(DXV*LOG2PI + scal[3]) - 0.5f*(DZV*LOG2PI + scal[2])
                 + 0.5f*(DZV*LOG2PI) - 0.5f*scal[0] + 0.5f*(float)DZV - 0.5f*trm;
    }
    kalman(y);                                            // -> fm, fc (new)

    // B1: vector history updates with OLD matrices
    for (int g = tid; g < t*16; g += 256){
      int s = g >> 4, i2 = g & 15;
      const float* Ar = transA + s*256 + i2*16;
      const float* Or = obsA   + s*256 + i2*16;
      float a = 0.f, b = 0.f;
      for (int k = 0; k < 16; ++k){ a += Ar[k]*ba_v[k]; b += Or[k]*ba_v[k]; }
      transB[s*16+i2] += a;
      obsB[s*16+i2]   += b;
    }
    __syncthreads();

    // B2: WMMA traces + matrix updates (wave-parallel over s < t)
    { v2f ctB[4]; load_bfrag_s(ctB, ct_m, lane, 16);
      for (int s = wave; s < t; s += 8){
        acc += wave_process(transA + s*256, (s==0)?Om0:OmT, bcov, ctB, stg, lane, false, true);
        acc += wave_process(obsA   + s*256, OmO,            bcov, ctB, stg, lane, false, true);
      }
    }
    __syncthreads();

    // new history entries at index t
    { int i = tid>>4, j = tid&15;
      float a = 0.f;
      for (int k = 0; k < 16; ++k) a += pW[i*16+k]*bA_m[k*16+j];
      transA[t*256+tid] = a - ((i==j)?1.f:0.f);
      obsA[t*256+tid]   = pH[tid]; }
    if (tid < 16){
      float a = 0.f;
      for (int k = 0; k < 16; ++k) a += pW[tid*16+k]*ba_v[k];
      transB[t*16+tid] = a + pb[tid];
      obsB[t*16+tid]   = peb[tid] - y[tid];
    }
    __syncthreads();
  }

  // -------- finalization --------
  if (tid == 0) scal[0] = 0.f;
  __syncthreads();
  spd_invert(tmpA, fc, &scal[0], tid, aug);               // logdet(fc)
  if (tid == 0) scal[1] += 0.5f*(DZV*LOG2PI + scal[0]);   // -= cst(dz, det fc)
  __syncthreads();

  // final traces: trace(Om A fc A)  (note A_li contraction -> transposed dot side)
  { v2f ctB[4]; load_bfrag_s(ctB, ct_m, lane, 16);
    for (int s = wave; s < TT; s += 8){
      acc += wave_process(transA + s*256, (s==0)?Om0:OmT, fc, ctB, stg, lane, true, false);
      acc += wave_process(obsA   + s*256, OmO,            fc, ctB, stg, lane, true, false);
    }
  }
  __syncthreads();

  // final quadratic forms: one thread per t
  { int t0 = tid;
    const float* At = transA + t0*256;
    const float* Ot = obsA   + t0*256;
    float ctv[16], cov_[16];
    for (int i = 0; i < 16; ++i){
      float a = transB[t0*16+i];
      float b = obsB[t0*16+i];
      for (int k = 0; k < 16; ++k){ a += At[i*16+k]*fm[k]; b += Ot[i*16+k]*fm[k]; }
      ctv[i] = a; cov_[i] = b;
    }
    const float* om = (t0==0) ? Om0 : OmT;
    for (int i = 0; i < 16; ++i){
      float zi = 0.f, wi = 0.f;
      for (int j = 0; j < 16; ++j){ zi += om[i*16+j]*ctv[j]; wi += OmO[i*16+j]*cov_[j]; }
      acc += ctv[i]*zi + cov_[i]*wi;
    }
  }

  red[tid] = acc;
  __syncthreads();
  if (tid == 0){
    float s = scal[1] + 0.5f*(float)DZV;
    for (int i = 0; i < 256; ++i) s += red[i];
    out[0] = s;
  }
}

extern "C" void kernel_launch(void* const* d_in, const int* in_sizes, int n_in,
                              void* d_out, int out_size, void* d_ws, size_t ws_size,
                              hipStream_t stream){
  (void)in_sizes; (void)n_in; (void)out_size; (void)ws_size;
  lgq_kernel<<<1, 256, 0, stream>>>(
      (const float*)d_in[0],  (const float*)d_in[1],  (const float*)d_in[2],
      (const float*)d_in[3],  (const float*)d_in[4],  (const float*)d_in[5],
      (const float*)d_in[6],  (const float*)d_in[7],  (const float*)d_in[8],
      (const float*)d_in[9],  (const float*)d_in[10], (const float*)d_in[11],
      (const float*)d_in[12], (const float*)d_in[13], (const float*)d_in[14],
      (const float*)d_in[15], (const float*)d_in[16],
      (float*)d_out, (float*)d_ws);
}